// LlamaAttention_15075335209654
// MI455X (gfx1250) — compile-verified
//
#include <hip/hip_runtime.h>
#include <hip/hip_bf16.h>
#include <stdint.h>

// ---------------------------------------------------------------------------
// Llama decode attention layer for gfx1250 (MI455X), f32, WMMA 16x16x4,
// async-to-LDS double-buffered weight streaming.
// B=32, HID=4096, H=32, D=128, BLK=16, MAXB=64, NBLK=2048, S=1024
// ---------------------------------------------------------------------------

#define B_     32
#define HID_   4096
#define H_     32
#define D_     128
#define BLK_   16
#define MAXB_  64
#define S_     1024

#define KCH    4               // split-K factor
#define KCHUNK (HID_ / KCH)    // 1024
#define SUB    64              // K sub-slice staged per async stage
#define LSTRIDE (SUB + 4)      // 68 -> banks (4*m)%64 distinct, conflict-free

typedef __attribute__((ext_vector_type(2))) float v2f;
typedef __attribute__((ext_vector_type(8))) float v8f;

// Async copy of 16 bytes global -> LDS (tracked by ASYNCcnt, no VGPR landing).
// LDS byte offset = low 32 bits of the generic shared-aperture address.
__device__ __forceinline__ void async_cp_b128(const float* __restrict__ g,
                                              float* l) {
  uint32_t loff = (uint32_t)(uintptr_t)l;
  uint64_t ga   = (uint64_t)(uintptr_t)g;
  asm volatile("global_load_async_to_lds_b128 %0, %1, off"
               :: "v"(loff), "v"(ga)
               : "memory");
}

// ---------------------------------------------------------------------------
// C[32, NTOT] = A[32,4096] @ W[n,:]^T  (W row-major [N,4096]), split-K=4.
// Block = 4 waves; each wave owns a 32x16 strip (two 16x16 WMMA accumulators).
// Both A and the block's 64 W rows are staged into LDS with double-buffered
// async-to-LDS b128 copies (12 async instr/wave per stage; wait <=12 keeps
// the next stage in flight while computing the current one).
// ---------------------------------------------------------------------------
__global__ void gemm_wmma(const float* __restrict__ A,
                          const float* __restrict__ W0,
                          const float* __restrict__ W1,
                          const float* __restrict__ W2,
                          float* __restrict__ part, int NTOT) {
  __shared__ float Abuf[2][32 * LSTRIDE];   // 2 * 8,704 B
  __shared__ float Wbuf[2][64 * LSTRIDE];   // 2 * 17,408 B   (52 KB total)

  const int tid  = threadIdx.x;             // 128 threads = 4 waves
  const int lane = tid & 31;
  const int wv   = tid >> 5;
  const int kc   = blockIdx.y;
  const int k0   = kc * KCHUNK;

  // block covers 64 consecutive output columns (never straddles a weight mat)
  const int nblk = blockIdx.x * 64;
  const float* W = (nblk < 4096) ? W0 : ((nblk < 8192) ? W1 : W2);
  const float* Wb = W + (size_t)(nblk & 4095) * HID_;   // 64 rows from here

  const int m  = lane & 15;
  const int hi = lane >> 4;                 // lanes 16-31 hold K+2,K+3

  // ---- async stage of one K sub-slice into buffer `buf` -------------------
  auto stage = [&](int buf, int k) {
    // A: 32 rows x 64 floats = 512 float4 -> 4 per thread
    #pragma unroll
    for (int i = 0; i < 4; i++) {
      int idx = tid + i * 128;
      int mm = idx >> 4, c4 = idx & 15;
      async_cp_b128(A + (size_t)mm * HID_ + k + c4 * 4,
                    &Abuf[buf][mm * LSTRIDE + c4 * 4]);
    }
    // W: 64 rows x 64 floats = 1024 float4 -> 8 per thread
    #pragma unroll
    for (int i = 0; i < 8; i++) {
      int idx = tid + i * 128;
      int r = idx >> 4, c4 = idx & 15;
      async_cp_b128(Wb + (size_t)r * HID_ + k + c4 * 4,
                    &Wbuf[buf][r * LSTRIDE + c4 * 4]);
    }
  };

  v8f c0 = {}; v8f c1 = {};

  stage(0, k0);
  int cur = 0;
  for (int ks = 0; ks < KCHUNK; ks += SUB) {
    const bool more = (ks + SUB) < KCHUNK;
    if (more) stage(cur ^ 1, k0 + ks + SUB);        // keep next tile in flight
    if (more) asm volatile("s_wait_asynccnt 0xc" ::: "memory");  // cur done
    else      asm volatile("s_wait_asynccnt 0x0" ::: "memory");
    __syncthreads();

    const float* a0p = &Abuf[cur][m * LSTRIDE + 2 * hi];
    const float* a1p = &Abuf[cur][(m + 16) * LSTRIDE + 2 * hi];
    const float* bp  = &Wbuf[cur][(wv * 16 + m) * LSTRIDE + 2 * hi];

    #pragma unroll
    for (int kk = 0; kk < SUB; kk += 4) {
      float2 av0 = *(const float2*)(a0p + kk);
      float2 av1 = *(const float2*)(a1p + kk);
      float2 bv  = *(const float2*)(bp + kk);
      v2f a0; a0[0] = av0.x; a0[1] = av0.y;
      v2f a1; a1[0] = av1.x; a1[1] = av1.y;
      v2f bf; bf[0] = bv.x;  bf[1] = bv.y;
      c0 = __builtin_amdgcn_wmma_f32_16x16x4_f32(false, a0, false, bf,
                                                 (short)0, c0, false, false);
      c1 = __builtin_amdgcn_wmma_f32_16x16x4_f32(false, a1, false, bf,
                                                 (short)0, c1, false, false);
    }
    __syncthreads();          // all waves done reading `cur` before re-stage
    cur ^= 1;
  }

  // C/D layout: VGPR i, lanes 0-15 -> M=i, lanes 16-31 -> M=i+8 ; N = lane&15
  const int n  = nblk % 4096 + wv * 16 + (lane & 15) + (nblk & ~4095);
  const int mb = hi * 8;
  float* pbase = part + (size_t)kc * 32 * NTOT;
  #pragma unroll
  for (int i = 0; i < 8; i++) {
    pbase[(size_t)(mb + i) * NTOT + n]      = c0[i];
    pbase[(size_t)(16 + mb + i) * NTOT + n] = c1[i];
  }
}

// ---------------------------------------------------------------------------
// Split-K reduce of fused QKV partials + RoPE(q,k).  One block per (b,h).
// part layout: [KCH][32][12288] with n<4096:q, <8192:k, else v.
// ---------------------------------------------------------------------------
__global__ void reduce_rope(const float* __restrict__ part,
                            const int* __restrict__ positions,
                            float* __restrict__ q_ws,
                            float* __restrict__ k_new,
                            float* __restrict__ v_new) {
  const int bh = blockIdx.x;
  const int b = bh >> 5, h = bh & 31;
  const int t = threadIdx.x;                       // 128
  const size_t NTOT = 3 * HID_;
  const size_t base = (size_t)b * NTOT + (size_t)h * D_ + t;

  float q = 0.f, k = 0.f, v = 0.f;
  #pragma unroll
  for (int kc = 0; kc < KCH; kc++) {
    const float* p = part + (size_t)kc * 32 * NTOT;
    q += p[base];
    k += p[base + 4096];
    v += p[base + 8192];
  }

  __shared__ float sq[D_], sk[D_];
  sq[t] = q; sk[t] = k;
  __syncthreads();

  const size_t ob = (size_t)b * HID_ + (size_t)h * D_;
  v_new[ob + t] = v;

  if (t < 64) {
    float pos  = (float)positions[b];
    float freq = pos * powf(10000.0f, -(float)t * (1.0f / 64.0f));
    float c = cosf(freq), s = sinf(freq);
    float q1 = sq[t], q2 = sq[t + 64];
    float k1 = sk[t], k2 = sk[t + 64];
    q_ws[ob + t]      = q1 * c - q2 * s;
    q_ws[ob + t + 64] = q1 * s + q2 * c;
    k_new[ob + t]      = k1 * c - k2 * s;
    k_new[ob + t + 64] = k1 * s + k2 * c;
  }
}

// ---------------------------------------------------------------------------
// Paged attention, one block (8 waves) per (b,h).  Wave-per-position dot
// products with b128 KV gathers; the freshly-written K/V row (position ctx-1)
// is sourced from workspace so inputs are never mutated.
// ---------------------------------------------------------------------------
__global__ void attn_kernel(const float* __restrict__ q_ws,
                            const float* __restrict__ k_new,
                            const float* __restrict__ v_new,
                            const float* __restrict__ k_cache,
                            const float* __restrict__ v_cache,
                            const int* __restrict__ block_tables,
                            const int* __restrict__ context_lens,
                            float* __restrict__ attn_out) {
  __shared__ float sc[S_];          // scores / probs
  __shared__ float red[8];
  __shared__ float accbuf[8 * D_];

  const int bh = blockIdx.x;
  const int b = bh >> 5, h = bh & 31;
  const int tid  = threadIdx.x;     // 256
  const int lane = tid & 31;
  const int wv   = tid >> 5;
  const int ctx  = context_lens[b];
  const int* bt  = block_tables + b * MAXB_;
  const size_t hb = (size_t)b * HID_ + (size_t)h * D_;
  const float scale = 0.08838834764831845f;   // 1/sqrt(128)

  const float4 qv = *(const float4*)(q_ws + hb + lane * 4);

  // ---- pass 1: scores -----------------------------------------------------
  for (int s = wv; s < ctx; s += 8) {
    const float* kp;
    if (s == ctx - 1) {
      kp = k_new + hb;
    } else {
      int blk = bt[s >> 4];
      kp = k_cache + ((((size_t)blk * BLK_ + (s & 15)) * H_ + h) * D_);
    }
    float4 kv = *(const float4*)(kp + lane * 4);
    float d = qv.x * kv.x + qv.y * kv.y + qv.z * kv.z + qv.w * kv.w;
    #pragma unroll
    for (int off = 16; off > 0; off >>= 1) d += __shfl_xor(d, off, 32);
    if (lane == 0) sc[s] = d * scale;
  }
  __syncthreads();

  // ---- softmax: max -------------------------------------------------------
  float m = -3.0e38f;
  for (int s = tid; s < ctx; s += 256) m = fmaxf(m, sc[s]);
  #pragma unroll
  for (int off = 16; off > 0; off >>= 1) m = fmaxf(m, __shfl_xor(m, off, 32));
  if (lane == 0) red[wv] = m;
  __syncthreads();
  m = red[0];
  #pragma unroll
  for (int i = 1; i < 8; i++) m = fmaxf(m, red[i]);

  // ---- softmax: exp + sum -------------------------------------------------
  float sum = 0.f;
  for (int s = tid; s < ctx; s += 256) {
    float e = __expf(sc[s] - m);
    sc[s] = e;
    sum += e;
  }
  #pragma unroll
  for (int off = 16; off > 0; off >>= 1) sum += __shfl_xor(sum, off, 32);
  __syncthreads();                       // red reads done; sc visible
  if (lane == 0) red[wv] = sum;
  __syncthreads();
  float denom = 0.f;
  #pragma unroll
  for (int i = 0; i < 8; i++) denom += red[i];
  const float inv = 1.0f / denom;

  // ---- pass 2: probs @ V --------------------------------------------------
  float4 acc = make_float4(0.f, 0.f, 0.f, 0.f);
  for (int s = wv; s < ctx; s += 8) {
    const float* vp;
    if (s == ctx - 1) {
      vp = v_new + hb;
    } else {
      int blk = bt[s >> 4];
      vp = v_cache + ((((size_t)blk * BLK_ + (s & 15)) * H_ + h) * D_);
    }
    float4 vv = *(const float4*)(vp + lane * 4);
    float p = sc[s];
    acc.x += p * vv.x; acc.y += p * vv.y;
    acc.z += p * vv.z; acc.w += p * vv.w;
  }
  *(float4*)(accbuf + wv * D_ + lane * 4) = acc;
  __syncthreads();

  if (tid < D_) {
    float r = 0.f;
    #pragma unroll
    for (int ww = 0; ww < 8; ww++) r += accbuf[ww * D_ + tid];
    attn_out[hb + tid] = r * inv;
  }
}

// ---------------------------------------------------------------------------
// Split-K reduce for the output projection -> d_out
// ---------------------------------------------------------------------------
__global__ void finalize(const float* __restrict__ part, float* __restrict__ out) {
  const int i = blockIdx.x * 256 + threadIdx.x;    // 131072 total
  float r = 0.f;
  #pragma unroll
  for (int kc = 0; kc < KCH; kc++) r += part[(size_t)kc * (32 * HID_) + i];
  out[i] = r;
}

// ---------------------------------------------------------------------------
extern "C" void kernel_launch(void* const* d_in, const int* in_sizes, int n_in,
                              void* d_out, int out_size, void* d_ws, size_t ws_size,
                              hipStream_t stream) {
  const float* hidden       = (const float*)d_in[0];
  const float* k_cache      = (const float*)d_in[1];
  const float* v_cache      = (const float*)d_in[2];
  const float* Wq           = (const float*)d_in[3];
  const float* Wk           = (const float*)d_in[4];
  const float* Wv           = (const float*)d_in[5];
  const float* Wo           = (const float*)d_in[6];
  const int*   positions    = (const int*)d_in[7];
  // d_in[8] = slot_mapping (redundant: slot == position ctx-1 of own batch)
  const int*   block_tables = (const int*)d_in[9];
  const int*   context_lens = (const int*)d_in[10];
  float*       out          = (float*)d_out;

  // workspace carve-up (floats)
  float* ws        = (float*)d_ws;
  float* qkv_part  = ws;                                   // 4*32*12288
  float* q_ws      = qkv_part + (size_t)KCH * 32 * (3 * HID_);
  float* k_new     = q_ws  + (size_t)B_ * HID_;
  float* v_new     = k_new + (size_t)B_ * HID_;
  float* attn_ws   = v_new + (size_t)B_ * HID_;
  float* out_part  = attn_ws + (size_t)B_ * HID_;          // 4*32*4096

  // 1) fused QKV projection:  N = 3*4096, 64 columns per block
  gemm_wmma<<<dim3(192, KCH), 128, 0, stream>>>(hidden, Wq, Wk, Wv,
                                                qkv_part, 3 * HID_);

  // 2) split-K reduce + RoPE -> q_ws, k_new, v_new
  reduce_rope<<<B_ * H_, D_, 0, stream>>>(qkv_part, positions,
                                          q_ws, k_new, v_new);

  // 3) paged attention
  attn_kernel<<<B_ * H_, 256, 0, stream>>>(q_ws, k_new, v_new,
                                           k_cache, v_cache,
                                           block_tables, context_lens,
                                           attn_ws);

  // 4) output projection:  N = 4096, 64 columns per block
  gemm_wmma<<<dim3(64, KCH), 128, 0, stream>>>(attn_ws, Wo, Wo, Wo,
                                               out_part, HID_);

  // 5) split-K reduce -> out
  finalize<<<(B_ * HID_) / 256, 256, 0, stream>>>(out_part, out);
}